// GATPolicy_42657615184429
// MI455X (gfx1250) — compile-verified
//
#include <hip/hip_runtime.h>
#include <hip/hip_bf16.h>

#define N_NODES 20000
#define N_EDGES 640000
#define E_TOT   (N_EDGES + N_NODES)
#define LAYERS  8
#define DDIM    64
#define HEADS   8

typedef __attribute__((ext_vector_type(2)))  float    v2f;
typedef __attribute__((ext_vector_type(8)))  float    v8f;

// ---------------------------------------------------------------------------
// float atomic max via sign-split integer atomics (works for mixed signs)
// ---------------------------------------------------------------------------
__device__ __forceinline__ void atomicMaxF(float* addr, float v) {
    if (v >= 0.0f) atomicMax((int*)addr, __float_as_int(v));
    else           atomicMin((unsigned int*)addr, __float_as_uint(v));
}

__device__ __forceinline__ void edge_sd(const long long* __restrict__ ei,
                                        int e, int& s, int& d) {
    if (e < N_EDGES) {
        s = (int)ei[e];
        d = (int)ei[(size_t)N_EDGES + e];
    } else {           // appended self-loops
        s = e - N_EDGES;
        d = s;
    }
}

// ---------------------------------------------------------------------------
// Kernel 1: h = x @ W via V_WMMA_F32_16X16X4_F32 (full fp32).
// One wave -> 16 rows x 64 cols tile; K=64 in 16 steps of 4.
// W staged in LDS with k-paired layout: Wl[(k>>1)*128 + col*2 + (k&1)]
// so each B fragment (W[k][col], W[k+1][col]) is ONE aligned ds_load_b64
// straight into an even VGPR pair -> no repacking movs in the inner loop.
// ---------------------------------------------------------------------------
__global__ __launch_bounds__(256) void gat_gemm(const float* __restrict__ x,
                                                const float* __restrict__ W,
                                                float* __restrict__ h) {
    __shared__ float Wl[DDIM * DDIM];
    int t = threadIdx.x;
    // preload + reorder into k-paired layout (16 elements per thread)
    for (int i = t; i < DDIM * DDIM; i += 256) {
        int k = i >> 6, col = i & 63;
        Wl[(k >> 1) * 128 + col * 2 + (k & 1)] = W[i];
    }
    __syncthreads();

    int wave = t >> 5, lane = t & 31;
    int m0   = (blockIdx.x * 8 + wave) * 16;
    int lrow = lane & 15;
    int row  = m0 + lrow;
    int srow = row < N_NODES ? row : N_NODES - 1;   // clamp; EXEC must stay all-1
    int kb   = (lane >> 4) * 2;                     // f32 A-frag: lanes16-31 hold K+2
    const float* xr = x + (size_t)srow * DDIM;

    v8f acc[4] = {};
    for (int k0 = 0; k0 < DDIM; k0 += 4) {
        v2f a = *(const v2f*)(xr + k0 + kb);        // global_load_b64, contiguous
        int p = (k0 + kb) >> 1;                     // k0+kb is even
#pragma unroll
        for (int nt = 0; nt < 4; ++nt) {
            v2f bf = *(const v2f*)&Wl[p * 128 + (nt * 16 + lrow) * 2];  // ds_load_b64
            acc[nt] = __builtin_amdgcn_wmma_f32_16x16x4_f32(
                false, a, false, bf, (short)0, acc[nt], false, false);
        }
    }

    int mhi = (lane >> 4) * 8;                      // C/D: lanes16-31 hold M=8..15
#pragma unroll
    for (int nt = 0; nt < 4; ++nt)
#pragma unroll
        for (int j = 0; j < 8; ++j) {
            int r = m0 + mhi + j;
            if (r < N_NODES) h[(size_t)r * DDIM + nt * 16 + lrow] = acc[nt][j];
        }
}

// ---------------------------------------------------------------------------
// Kernel 2: per-(node,head) attention scores + init m/z/acc
// ---------------------------------------------------------------------------
__global__ void gat_scores(const float* __restrict__ h,
                           const float* __restrict__ a_src,
                           const float* __restrict__ a_dst,
                           float* __restrict__ s_src, float* __restrict__ s_dst,
                           float* __restrict__ mbuf, float* __restrict__ zbuf,
                           float* __restrict__ acc) {
    int i = blockIdx.x * blockDim.x + threadIdx.x;
    if (i >= N_NODES * HEADS) return;
    int n = i >> 3, hd = i & 7;
    const float* hp = h + (size_t)n * DDIM + hd * 8;
    float ss = 0.f, sd = 0.f;
#pragma unroll
    for (int c = 0; c < 8; ++c) {
        float v = hp[c];
        ss += v * a_src[hd * 8 + c];
        sd += v * a_dst[hd * 8 + c];
    }
    s_src[i] = ss;
    s_dst[i] = sd;
    mbuf[i]  = -3.0e38f;
    zbuf[i]  = 0.0f;
    float* ap = acc + (size_t)n * DDIM + hd * 8;
#pragma unroll
    for (int c = 0; c < 8; ++c) ap[c] = 0.0f;
}

// ---------------------------------------------------------------------------
// Kernel 3: per-edge leaky-relu score, segment max into mbuf
// ---------------------------------------------------------------------------
__global__ void gat_edge_max(const long long* __restrict__ ei,
                             const float* __restrict__ s_src,
                             const float* __restrict__ s_dst,
                             float* __restrict__ mbuf) {
    int e = blockIdx.x * blockDim.x + threadIdx.x;
    if (e >= E_TOT) return;
    int s, d;
    edge_sd(ei, e, s, d);
#pragma unroll
    for (int hd = 0; hd < 8; ++hd) {
        float v = s_src[(size_t)s * 8 + hd] + s_dst[(size_t)d * 8 + hd];
        v = v > 0.0f ? v : 0.2f * v;
        atomicMaxF(&mbuf[(size_t)d * 8 + hd], v);
    }
}

// ---------------------------------------------------------------------------
// Kernel 4: exp(e - m[dst]) -> exw, segment sum into zbuf
// ---------------------------------------------------------------------------
__global__ void gat_edge_exp(const long long* __restrict__ ei,
                             const float* __restrict__ s_src,
                             const float* __restrict__ s_dst,
                             const float* __restrict__ mbuf,
                             float* __restrict__ zbuf,
                             float* __restrict__ exw) {
    int e = blockIdx.x * blockDim.x + threadIdx.x;
    if (e >= E_TOT) return;
    int s, d;
    edge_sd(ei, e, s, d);
#pragma unroll
    for (int hd = 0; hd < 8; ++hd) {
        float v = s_src[(size_t)s * 8 + hd] + s_dst[(size_t)d * 8 + hd];
        v = v > 0.0f ? v : 0.2f * v;
        float ex = __expf(v - mbuf[(size_t)d * 8 + hd]);
        exw[(size_t)e * 8 + hd] = ex;
        atomicAdd(&zbuf[(size_t)d * 8 + hd], ex);
    }
}

// ---------------------------------------------------------------------------
// Kernel 5: acc[dst] += h[src] * exw ; 16 threads per edge, float4 each
// ---------------------------------------------------------------------------
__global__ void gat_scatter(const long long* __restrict__ ei,
                            const float* __restrict__ h,
                            const float* __restrict__ exw,
                            float* __restrict__ acc) {
    int idx = blockIdx.x * blockDim.x + threadIdx.x;
    if (idx >= E_TOT * 16) return;
    int e = idx >> 4, q = idx & 15;   // q*4 = starting channel, head = q>>1
    int s, d;
    edge_sd(ei, e, s, d);
    float a = exw[(size_t)e * 8 + (q >> 1)];
    const float4 hv = *(const float4*)(h + (size_t)s * DDIM + q * 4);
    float* op = acc + (size_t)d * DDIM + q * 4;
    atomicAdd(op + 0, hv.x * a);
    atomicAdd(op + 1, hv.y * a);
    atomicAdd(op + 2, hv.z * a);
    atomicAdd(op + 3, hv.w * a);
}

// ---------------------------------------------------------------------------
// Kernel 6: normalize by z + 1e-16, add bias
// ---------------------------------------------------------------------------
__global__ void gat_finalize(const float* __restrict__ acc,
                             const float* __restrict__ zbuf,
                             const float* __restrict__ bias,
                             float* __restrict__ xout) {
    int i = blockIdx.x * blockDim.x + threadIdx.x;
    if (i >= N_NODES * DDIM) return;
    int n = i >> 6, dcol = i & 63;
    xout[i] = acc[i] / (zbuf[(size_t)n * 8 + (dcol >> 3)] + 1e-16f) + bias[dcol];
}

// ---------------------------------------------------------------------------
extern "C" void kernel_launch(void* const* d_in, const int* in_sizes, int n_in,
                              void* d_out, int out_size, void* d_ws, size_t ws_size,
                              hipStream_t stream) {
    (void)in_sizes; (void)n_in; (void)out_size; (void)ws_size;

    const float*     x    = (const float*)d_in[0];
    const long long* ei   = (const long long*)d_in[1];   // jnp.int64 edge_index [2,E]
    const float*     Wall = (const float*)d_in[2];       // [L,64,64]
    const float*     aS   = (const float*)d_in[3];       // [L,8,8]
    const float*     aD   = (const float*)d_in[4];       // [L,8,8]
    const float*     bAll = (const float*)d_in[5];       // [L,64]
    float* out = (float*)d_out;

    char* p = (char*)d_ws;
    auto take = [&](size_t bytes) -> void* {
        void* r = p;
        p += (bytes + 255) & ~(size_t)255;
        return r;
    };
    float* hbuf = (float*)take((size_t)N_NODES * DDIM  * 4);
    float* acc  = (float*)take((size_t)N_NODES * DDIM  * 4);
    float* xbuf = (float*)take((size_t)N_NODES * DDIM  * 4);
    float* ssrc = (float*)take((size_t)N_NODES * HEADS * 4);
    float* sdst = (float*)take((size_t)N_NODES * HEADS * 4);
    float* mbuf = (float*)take((size_t)N_NODES * HEADS * 4);
    float* zbuf = (float*)take((size_t)N_NODES * HEADS * 4);
    float* exw  = (float*)take((size_t)E_TOT   * HEADS * 4);

    dim3 blk(256);
    const int gGemm  = (N_NODES / 16 + 7) / 8;            // 157 blocks, 8 waves each
    const int gScore = (N_NODES * HEADS + 255) / 256;
    const int gEdge  = (E_TOT + 255) / 256;
    const int gScat  = (E_TOT * 16 + 255) / 256;
    const int gFin   = (N_NODES * DDIM + 255) / 256;

    for (int l = 0; l < LAYERS; ++l) {
        const float* xin  = (l == 0) ? x : xbuf;
        float*       xout = (l == LAYERS - 1) ? out : xbuf;

        gat_gemm    <<<gGemm,  blk, 0, stream>>>(xin, Wall + (size_t)l * DDIM * DDIM, hbuf);
        gat_scores  <<<gScore, blk, 0, stream>>>(hbuf, aS + l * 64, aD + l * 64,
                                                 ssrc, sdst, mbuf, zbuf, acc);
        gat_edge_max<<<gEdge,  blk, 0, stream>>>(ei, ssrc, sdst, mbuf);
        gat_edge_exp<<<gEdge,  blk, 0, stream>>>(ei, ssrc, sdst, mbuf, zbuf, exw);
        gat_scatter <<<gScat,  blk, 0, stream>>>(ei, hbuf, exw, acc);
        gat_finalize<<<gFin,   blk, 0, stream>>>(acc, zbuf, bAll + l * 64, xout);
    }
}